// TrajGenerator_39470749450879
// MI455X (gfx1250) — compile-verified
//
#include <hip/hip_runtime.h>
#include <math.h>

#define NAG   128   // N agents
#define HOR   25    // HORIZON == FUT_LEN
#define DM    64    // D_MODEL
#define G3    192   // 3*D_MODEL (GRU gates)
#define SC_C  32
#define NS_C  32
#define OG_C  32
#define KPAD  80    // padded K for the 66-wide concat features
#define GPIX  625   // 25*25

typedef __attribute__((ext_vector_type(2))) float v2f;
typedef __attribute__((ext_vector_type(8))) float v8f;

__device__ __forceinline__ float lrelu(float v) { return v > 0.f ? v : 0.1f * v; }
__device__ __forceinline__ float sigm(float v)  { return 1.f / (1.f + __expf(-v)); }

// ---------------- WMMA f32 16x16x4 helpers (CDNA5 layouts) ----------------
__device__ __forceinline__ v8f wmma_acc(v2f a, v2f b, v8f c) {
  return __builtin_amdgcn_wmma_f32_16x16x4_f32(false, a, false, b, (short)0, c,
                                               false, false);
}

// A fragment: 16x4 tile of row-major A, rows row0..row0+15, K cols k0..k0+3.
__device__ __forceinline__ v2f load_a(const float* A, int lda, int row0, int k0,
                                      int lane) {
  int mm = lane & 15, hl = lane >> 4;
  const float* p = A + (row0 + mm) * lda + k0 + hl * 2;
  v2f a; a.x = p[0]; a.y = p[1]; return a;
}

// B fragment: 4x16 tile B[k][n] = W[n0+n][k0+k] for row-major weight W(out,K).
__device__ __forceinline__ v2f load_b(const float* W, int ldw, int n0, int k0,
                                      int lane) {
  int nn = lane & 15, hl = lane >> 4;
  const float* p = W + (n0 + nn) * ldw + k0 + hl * 2;
  v2f b; b.x = p[0]; b.y = p[1]; return b;
}

// Guarded B fragment (K not multiple of 4 -> zero pad past kmax).
__device__ __forceinline__ v2f load_b_g(const float* W, int ldw, int n0, int k0,
                                        int kmax, int lane) {
  int nn = lane & 15, hl = lane >> 4;
  int k = k0 + hl * 2;
  v2f b;
  b.x = (k     < kmax) ? W[(n0 + nn) * ldw + k]     : 0.f;
  b.y = (k + 1 < kmax) ? W[(n0 + nn) * ldw + k + 1] : 0.f;
  return b;
}

// A fragment for the decoder concat [ip1(64) | lf_t(66, padded to 80)].
__device__ __forceinline__ v2f load_a_cat(const float* ip1, const float* lf,
                                          int row0, int t, int k0, int lane) {
  int mm = lane & 15, hl = lane >> 4;
  int row = row0 + mm;
  int k = k0 + hl * 2;
  v2f a;
  a.x = (k     < 64) ? ip1[row * DM + k]     : lf[(row * HOR + t) * KPAD + (k - 64)];
  a.y = (k + 1 < 64) ? ip1[row * DM + k + 1] : lf[(row * HOR + t) * KPAD + (k - 63)];
  return a;
}

// ---------------- bilinear grid sample (align_corners=False, clamped) ------
__device__ __forceinline__ void bilin(const float* img, int C, float gx, float gy,
                                      float* out) {
  float x = fminf(fmaxf((gx + 1.f) * 12.5f - 0.5f, 0.f), 24.f);
  float y = fminf(fmaxf((gy + 1.f) * 12.5f - 0.5f, 0.f), 24.f);
  float x0f = floorf(x), y0f = floorf(y);
  int x0 = (int)x0f, y0 = (int)y0f;
  int x1 = min(x0 + 1, 24), y1 = min(y0 + 1, 24);
  float wx = x - x0f, wy = y - y0f;
  float w00 = (1.f - wx) * (1.f - wy), w01 = wx * (1.f - wy);
  float w10 = (1.f - wx) * wy,         w11 = wx * wy;
  for (int c = 0; c < C; ++c) {
    const float* p = img + c * GPIX;
    out[c] = w00 * p[y0 * 25 + x0] + w01 * p[y0 * 25 + x1] +
             w10 * p[y1 * 25 + x0] + w11 * p[y1 * 25 + x1];
  }
}

// ---------------- K1: grid samples + rotations + concat features -----------
__global__ void k_prep(const float* __restrict__ waypts, const float* __restrict__ r_mat,
                       const float* __restrict__ scene,  const float* __restrict__ nsf,
                       const float* __restrict__ omg,    const float* __restrict__ futrot,
                       float* __restrict__ wcat, float* __restrict__ lf,
                       float* __restrict__ fpr) {
  int i = blockIdx.x * blockDim.x + threadIdx.x;
  if (i >= NAG * HOR) return;
  int n = i / HOR, t = i % HOR;
  const float* R = r_mat + n * 4;
  float r00 = R[0], r01 = R[1], r10 = R[2], r11 = R[3];
  float wx = waypts[i * 2], wy = waypts[i * 2 + 1];
  // encoder inputs: [waypts_rot(2) | scene(32) | ns(32)] padded to 80
  float* wc = wcat + i * KPAD;
  wc[0] = r00 * wx + r01 * wy;
  wc[1] = r10 * wx + r11 * wy;
  bilin(scene + n * SC_C * GPIX, SC_C, wx, wy, wc + 2);
  bilin(nsf + i * NS_C * GPIX, NS_C, wx, wy, wc + 2 + SC_C);
  for (int c = 66; c < KPAD; ++c) wc[c] = 0.f;
  // decoder inputs: fut_prev_rot, fut_prev = R^T*fpr/20, [scene|omg|fpr] -> 80
  float fprx, fpry;
  if (t == 0) {
    float w0x = waypts[n * HOR * 2], w0y = waypts[n * HOR * 2 + 1];
    fprx = r00 * w0x + r01 * w0y;
    fpry = r10 * w0x + r11 * w0y;
  } else {
    fprx = futrot[(n * HOR + t - 1) * 2];
    fpry = futrot[(n * HOR + t - 1) * 2 + 1];
  }
  float fpx = (r00 * fprx + r10 * fpry) * 0.05f;
  float fpy = (r01 * fprx + r11 * fpry) * 0.05f;
  float* l = lf + i * KPAD;
  bilin(scene + n * SC_C * GPIX, SC_C, fpx, fpy, l);
  bilin(omg + i * OG_C * GPIX, OG_C, fpx, fpy, l + SC_C);
  l[64] = fprx; l[65] = fpry;
  for (int c = 66; c < KPAD; ++c) l[c] = 0.f;
  fpr[i * 2] = fprx; fpr[i * 2 + 1] = fpry;
}

// ---------------- K2: motion embedding h0 = lrelu(motion@W^T+b) (128x64) ---
__global__ void k_hist(const float* __restrict__ motion, const float* __restrict__ W,
                       const float* __restrict__ b, float* __restrict__ h0) {
  int lane = threadIdx.x & 31, wave = threadIdx.x >> 5;
  int hl = lane >> 4, nn = lane & 15;
  for (int ti = 0; ti < 4; ++ti) {
    int tile = wave * 4 + ti;           // 32 tiles: 8 row x 4 col
    int rt = tile >> 2, ct = tile & 3;
    v8f c = {};
    for (int k = 0; k < 32; k += 4) {
      v2f a  = load_a(motion, 32, rt * 16, k, lane);
      v2f bb = load_b(W, 32, ct * 16, k, lane);
      c = wmma_acc(a, bb, c);
    }
    int col = ct * 16 + nn;
    for (int r = 0; r < 8; ++r)
      h0[(rt * 16 + r + hl * 8) * DM + col] = lrelu(c[r] + b[col]);
  }
}

// ---------------- K3: waypoint embedding (3200x66 -> 64), lrelu ------------
__global__ void k_wemb(const float* __restrict__ wcat, const float* __restrict__ W,
                       const float* __restrict__ b, float* __restrict__ wf) {
  int lane = threadIdx.x & 31, wave = threadIdx.x >> 5;
  int tile = blockIdx.x * 8 + wave;     // 800 tiles: 200 row x 4 col
  int rt = tile >> 2, ct = tile & 3;
  v8f c = {};
  for (int k = 0; k < 68; k += 4) {
    v2f a  = load_a(wcat, KPAD, rt * 16, k, lane);       // padded with zeros
    v2f bb = load_b_g(W, 66, ct * 16, k, 66, lane);
    c = wmma_acc(a, bb, c);
  }
  int col = ct * 16 + (lane & 15), hl = lane >> 4;
  for (int r = 0; r < 8; ++r)
    wf[(rt * 16 + r + hl * 8) * DM + col] = lrelu(c[r] + b[col]);
}

// ---------------- K4: encoder xg = wf @ Wih^T + bih (3200x192) -------------
__global__ void k_xg(const float* __restrict__ wf, const float* __restrict__ Wih,
                     const float* __restrict__ bih, float* __restrict__ xg) {
  int lane = threadIdx.x & 31, wave = threadIdx.x >> 5;
  int tile = blockIdx.x * 8 + wave;     // 2400 tiles: 200 row x 12 col
  int rt = tile / 12, ct = tile % 12;
  v8f c = {};
  for (int k = 0; k < 64; k += 4) {
    v2f a  = load_a(wf, DM, rt * 16, k, lane);
    v2f bb = load_b(Wih, DM, ct * 16, k, lane);
    c = wmma_acc(a, bb, c);
  }
  int col = ct * 16 + (lane & 15), hl = lane >> 4;
  for (int r = 0; r < 8; ++r)
    xg[(rt * 16 + r + hl * 8) * G3 + col] = c[r] + bih[col];
}

// ---------------- K5: persistent encoder GRU (sequential over T=25) --------
__global__ void k_enc(const float* __restrict__ xg, const float* __restrict__ Whh,
                      const float* __restrict__ bhh, const int* __restrict__ lens,
                      float* __restrict__ hwy) {
  __shared__ float hsh[NAG * DM];
  int tid = threadIdx.x;
  for (int i = tid; i < NAG * DM; i += 256) hsh[i] = 0.f;
  __syncthreads();
  int lane = tid & 31, wave = tid >> 5;
  int mm = lane & 15, hl = lane >> 4;
  for (int t = 0; t < HOR; ++t) {
    v8f cr[4], cz[4], cn[4];
    // h @ Whh^T : each task owns 16x16 hidden block -> r,z,n gate tiles in regs
    for (int ti = 0; ti < 4; ++ti) {
      int task = wave + ti * 8;         // 32 tasks: 8 row x 4 hidden-col
      int rt = task >> 2, ct = task & 3;
      v8f ar = {}, az = {}, an = {};
      for (int k = 0; k < 64; k += 4) {
        v2f a  = load_a(hsh, DM, rt * 16, k, lane);
        v2f br = load_b(Whh, DM, ct * 16, k, lane);
        v2f bz = load_b(Whh, DM, 64 + ct * 16, k, lane);
        v2f bn = load_b(Whh, DM, 128 + ct * 16, k, lane);
        ar = wmma_acc(a, br, ar);
        az = wmma_acc(a, bz, az);
        an = wmma_acc(a, bn, an);
      }
      cr[ti] = ar; cz[ti] = az; cn[ti] = an;
    }
    __syncthreads();   // all reads of hsh done before update
    for (int ti = 0; ti < 4; ++ti) {
      int task = wave + ti * 8;
      int rt = task >> 2, ct = task & 3;
      int col = ct * 16 + mm;
      for (int r = 0; r < 8; ++r) {
        int ag = rt * 16 + r + hl * 8;
        int xrow = (ag * HOR + t) * G3;
        float hr = cr[ti][r] + bhh[col];
        float hz = cz[ti][r] + bhh[64 + col];
        float hn = cn[ti][r] + bhh[128 + col];
        float rr = sigm(xg[xrow + col] + hr);
        float zz = sigm(xg[xrow + 64 + col] + hz);
        float nv = tanhf(xg[xrow + 128 + col] + rr * hn);
        float hold = hsh[ag * DM + col];
        float hnew = (1.f - zz) * nv + zz * hold;
        bool v = t < lens[ag];
        hsh[ag * DM + col] = v ? hnew : hold;
        hwy[(ag * HOR + t) * DM + col] = v ? hnew : 0.f;
      }
    }
    __syncthreads();
  }
}

// ---------------- K6: key/value projections (3200x64 each) -----------------
__global__ void k_kv(const float* __restrict__ hwy,
                     const float* __restrict__ Wk, const float* __restrict__ bk,
                     const float* __restrict__ Wv, const float* __restrict__ bv,
                     float* __restrict__ key, float* __restrict__ val) {
  int lane = threadIdx.x & 31, wave = threadIdx.x >> 5;
  int id = blockIdx.x * 8 + wave;       // 1600 tiles (800 key + 800 value)
  int which = id >= 800;
  int tile = which ? id - 800 : id;
  int rt = tile >> 2, ct = tile & 3;
  const float* W = which ? Wv : Wk;
  const float* b = which ? bv : bk;
  float* out = which ? val : key;
  v8f c = {};
  for (int k = 0; k < 64; k += 4) {
    v2f a  = load_a(hwy, DM, rt * 16, k, lane);
    v2f bb = load_b(W, DM, ct * 16, k, lane);
    c = wmma_acc(a, bb, c);
  }
  int col = ct * 16 + (lane & 15), hl = lane >> 4;
  for (int r = 0; r < 8; ++r)
    out[(rt * 16 + r + hl * 8) * DM + col] = c[r] + b[col];
}

// ---------------- K7: persistent decoder (sequential over 25 steps) --------
__global__ void k_dec(const float* __restrict__ h0, const float* __restrict__ key,
                      const float* __restrict__ val, const float* __restrict__ lf,
                      const float* __restrict__ fpr, const int* __restrict__ lens,
                      const float* __restrict__ Wq, const float* __restrict__ bq,
                      const float* __restrict__ attW, const float* __restrict__ attb,
                      const float* __restrict__ Wih, const float* __restrict__ bih,
                      const float* __restrict__ Whh, const float* __restrict__ bhh,
                      const float* __restrict__ opW, const float* __restrict__ opb,
                      float* __restrict__ qb, float* __restrict__ ip1,
                      float* __restrict__ ipb, float* __restrict__ out) {
  __shared__ float hsh[NAG * DM];
  int tid = threadIdx.x;
  for (int i = tid; i < NAG * DM; i += 256) hsh[i] = h0[i];
  __syncthreads();
  int lane = tid & 31, wave = tid >> 5;
  int mm = lane & 15, hl = lane >> 4;
  for (int t = 0; t < HOR; ++t) {
    // Phase A: q = h @ Wq^T + bq (128x64)
    for (int ti = 0; ti < 4; ++ti) {
      int tile = wave * 4 + ti;
      int rt = tile >> 2, ct = tile & 3;
      v8f c = {};
      for (int k = 0; k < 64; k += 4) {
        v2f a  = load_a(hsh, DM, rt * 16, k, lane);
        v2f bb = load_b(Wq, DM, ct * 16, k, lane);
        c = wmma_acc(a, bb, c);
      }
      int col = ct * 16 + mm;
      for (int r = 0; r < 8; ++r)
        qb[(rt * 16 + r + hl * 8) * DM + col] = c[r] + bq[col];
    }
    __threadfence(); __syncthreads();
    // Phase B: masked softmax attention per (agent, head), D_K=16, T=25
    for (int item = tid; item < NAG * 4; item += 256) {
      int ag = item >> 2, hd = item & 3;
      int base = hd * 16;
      float qv[16];
#pragma unroll
      for (int j = 0; j < 16; ++j) qv[j] = qb[ag * DM + base + j];
      int L = lens[ag];
      float s[HOR];
      float mx = -1.0e9f;
#pragma unroll
      for (int tt = 0; tt < HOR; ++tt) {
        const float* kp = key + (ag * HOR + tt) * DM + base;
        float d = 0.f;
#pragma unroll
        for (int j = 0; j < 16; ++j) d += qv[j] * kp[j];
        s[tt] = (tt < L) ? d * 0.25f : -1.0e9f;
        mx = fmaxf(mx, s[tt]);
      }
      float den = 0.f;
#pragma unroll
      for (int tt = 0; tt < HOR; ++tt) { s[tt] = __expf(s[tt] - mx); den += s[tt]; }
      float inv = 1.f / den;
      float acc[16];
#pragma unroll
      for (int j = 0; j < 16; ++j) acc[j] = 0.f;
#pragma unroll
      for (int tt = 0; tt < HOR; ++tt) {
        const float* vp = val + (ag * HOR + tt) * DM + base;
        float w = s[tt];
#pragma unroll
        for (int j = 0; j < 16; ++j) acc[j] += w * vp[j];
      }
#pragma unroll
      for (int j = 0; j < 16; ++j) ip1[ag * DM + base + j] = acc[j] * inv;
    }
    __threadfence(); __syncthreads();
    // Phase C: ip = lrelu([ip1 | lf_t] @ attW^T + attb)  (K=130 padded to 132)
    for (int ti = 0; ti < 4; ++ti) {
      int tile = wave * 4 + ti;
      int rt = tile >> 2, ct = tile & 3;
      v8f c = {};
      for (int k = 0; k < 132; k += 4) {
        v2f a  = load_a_cat(ip1, lf, rt * 16, t, k, lane);
        v2f bb = load_b_g(attW, 130, ct * 16, k, 130, lane);
        c = wmma_acc(a, bb, c);
      }
      int col = ct * 16 + mm;
      for (int r = 0; r < 8; ++r)
        ipb[(rt * 16 + r + hl * 8) * DM + col] = lrelu(c[r] + attb[col]);
    }
    __threadfence(); __syncthreads();
    // Phase D: GRU cell — 6 gate tiles per task fully in registers
    {
      v8f cxr[4], cxz[4], cxn[4], chrr[4], chz[4], chn[4];
      for (int ti = 0; ti < 4; ++ti) {
        int task = wave + ti * 8;
        int rt = task >> 2, ct = task & 3;
        v8f xr = {}, xz = {}, xn = {}, hr = {}, hz = {}, hn = {};
        for (int k = 0; k < 64; k += 4) {
          v2f ai = load_a(ipb, DM, rt * 16, k, lane);
          v2f ah = load_a(hsh, DM, rt * 16, k, lane);
          v2f br = load_b(Wih, DM, ct * 16, k, lane);
          v2f bz = load_b(Wih, DM, 64 + ct * 16, k, lane);
          v2f bn = load_b(Wih, DM, 128 + ct * 16, k, lane);
          xr = wmma_acc(ai, br, xr);
          xz = wmma_acc(ai, bz, xz);
          xn = wmma_acc(ai, bn, xn);
          v2f hbr = load_b(Whh, DM, ct * 16, k, lane);
          v2f hbz = load_b(Whh, DM, 64 + ct * 16, k, lane);
          v2f hbn = load_b(Whh, DM, 128 + ct * 16, k, lane);
          hr = wmma_acc(ah, hbr, hr);
          hz = wmma_acc(ah, hbz, hz);
          hn = wmma_acc(ah, hbn, hn);
        }
        cxr[ti] = xr; cxz[ti] = xz; cxn[ti] = xn;
        chrr[ti] = hr; chz[ti] = hz; chn[ti] = hn;
      }
      __syncthreads();
      for (int ti = 0; ti < 4; ++ti) {
        int task = wave + ti * 8;
        int rt = task >> 2, ct = task & 3;
        int col = ct * 16 + mm;
        for (int r = 0; r < 8; ++r) {
          int ag = rt * 16 + r + hl * 8;
          float rr = sigm(cxr[ti][r] + bih[col]       + chrr[ti][r] + bhh[col]);
          float zz = sigm(cxz[ti][r] + bih[64 + col]  + chz[ti][r]  + bhh[64 + col]);
          float nv = tanhf(cxn[ti][r] + bih[128 + col] +
                           rr * (chn[ti][r] + bhh[128 + col]));
          float hold = hsh[ag * DM + col];
          hsh[ag * DM + col] = (1.f - zz) * nv + zz * hold;
        }
      }
    }
    __syncthreads();
    // Phase E: output head (128x5) + residual/exp/tanh transform
    for (int i = tid; i < NAG * 5; i += 256) {
      int ag = i / 5, o = i % 5;
      float d = opb[o];
      const float* hw = hsh + ag * DM;
      const float* w = opW + o * DM;
#pragma unroll
      for (int j = 0; j < DM; ++j) d += hw[j] * w[j];
      float res;
      if (o < 2)      res = d + fpr[(ag * HOR + t) * 2 + o];
      else if (o < 4) res = __expf(d);
      else            res = tanhf(d);
      out[(ag * HOR + t) * 5 + o] = res;
    }
    __syncthreads();
  }
}

extern "C" void kernel_launch(void* const* d_in, const int* in_sizes, int n_in,
                              void* d_out, int out_size, void* d_ws, size_t ws_size,
                              hipStream_t stream) {
  (void)in_sizes; (void)n_in; (void)out_size; (void)ws_size;
  const float* motion  = (const float*)d_in[0];
  const float* scene   = (const float*)d_in[1];
  const float* nsf     = (const float*)d_in[2];
  const float* waypts  = (const float*)d_in[3];
  const int*   lens    = (const int*)  d_in[4];
  const float* r_mat   = (const float*)d_in[5];
  const float* omg     = (const float*)d_in[6];
  const float* futrot  = (const float*)d_in[7];
  const float* hist_W  = (const float*)d_in[8];
  const float* hist_b  = (const float*)d_in[9];
  const float* wemb_W  = (const float*)d_in[10];
  const float* wemb_b  = (const float*)d_in[11];
  const float* enc_Wih = (const float*)d_in[12];
  const float* enc_bih = (const float*)d_in[13];
  const float* enc_Whh = (const float*)d_in[14];
  const float* enc_bhh = (const float*)d_in[15];
  const float* att_W   = (const float*)d_in[16];
  const float* att_b   = (const float*)d_in[17];
  const float* Wk      = (const float*)d_in[18];
  const float* bk      = (const float*)d_in[19];
  const float* Wq      = (const float*)d_in[20];
  const float* bq      = (const float*)d_in[21];
  const float* Wv      = (const float*)d_in[22];
  const float* bv      = (const float*)d_in[23];
  const float* dec_Wih = (const float*)d_in[24];
  const float* dec_bih = (const float*)d_in[25];
  const float* dec_Whh = (const float*)d_in[26];
  const float* dec_bhh = (const float*)d_in[27];
  const float* op_W    = (const float*)d_in[28];
  const float* op_b    = (const float*)d_in[29];
  float* outp = (float*)d_out;

  float* ws = (float*)d_ws;
  const int NT = NAG * HOR;                 // 3200
  float* wcat = ws;                 ws += NT * KPAD;   // 256000
  float* lf   = ws;                 ws += NT * KPAD;   // 256000
  float* fpr  = ws;                 ws += NT * 2;      // 6400
  float* wf   = ws;                 ws += NT * DM;     // 204800
  float* xg   = ws;                 ws += NT * G3;     // 614400
  float* hwy  = ws;                 ws += NT * DM;     // 204800
  float* key  = ws;                 ws += NT * DM;     // 204800
  float* valv = ws;                 ws += NT * DM;     // 204800
  float* h0   = ws;                 ws += NAG * DM;    // 8192
  float* qb   = ws;                 ws += NAG * DM;    // 8192
  float* ip1  = ws;                 ws += NAG * DM;    // 8192
  float* ipb  = ws;                 ws += NAG * DM;    // 8192

  k_prep<<<(NT + 255) / 256, 256, 0, stream>>>(waypts, r_mat, scene, nsf, omg,
                                               futrot, wcat, lf, fpr);
  k_hist<<<1, 256, 0, stream>>>(motion, hist_W, hist_b, h0);
  k_wemb<<<100, 256, 0, stream>>>(wcat, wemb_W, wemb_b, wf);
  k_xg<<<300, 256, 0, stream>>>(wf, enc_Wih, enc_bih, xg);
  k_enc<<<1, 256, 0, stream>>>(xg, enc_Whh, enc_bhh, lens, hwy);
  k_kv<<<200, 256, 0, stream>>>(hwy, Wk, bk, Wv, bv, key, valv);
  k_dec<<<1, 256, 0, stream>>>(h0, key, valv, lf, fpr, lens, Wq, bq, att_W, att_b,
                               dec_Wih, dec_bih, dec_Whh, dec_bhh, op_W, op_b,
                               qb, ip1, ipb, outp);
}